// ImageNormalizer_8718783611218
// MI455X (gfx1250) — compile-verified
//
#include <hip/hip_runtime.h>
#include <cstdint>
#include <cstddef>

// ---------------------------------------------------------------------------
// ImageNormalizer for MI455X (gfx1250).
// Memory-bound (≈360MB traffic -> ~16us @ 23.3TB/s). Single fused kernel:
//  - async global->LDS staging of the stencil halo tile (CDNA5 ASYNCcnt path)
//  - fused double 9x9 box filter (separable, reflect padding) for CLAHE
//  - pointwise raw/log/asinh channels, b128 (float4) output stores
// ---------------------------------------------------------------------------

#define IN_H   512
#define IN_W   512
#define NIMG   40          // B*C = 8*5
#define NBANDS 5
#define NCH    7           // raw, log(t=0), log(t=1), clahe, asinh x3
#define NT     256         // threads per block (8 wave32)

#define TILE_H 32
#define TILE_W 64
#define XS_H   (TILE_H + 16)   // 48  (radius 8 halo: two chained radius-4 boxes)
#define XS_W   (TILE_W + 16)   // 80
#define HX_H   XS_H            // 48
#define HX_W   (TILE_W + 8)    // 72
#define RS_H   (TILE_H + 8)    // 40
#define RS_W   (TILE_W + 8)    // 72
#define HR_H   (TILE_H + 8)    // 40
#define HR_W   TILE_W          // 64

#define INV81  (1.0f / 81.0f)

#ifndef __has_builtin
#define __has_builtin(x) 0
#endif

#if __has_builtin(__builtin_amdgcn_global_load_async_to_lds_b32) && \
    __has_builtin(__builtin_amdgcn_s_wait_asynccnt)
#define USE_ASYNC_LDS 1
#else
#define USE_ASYNC_LDS 0
#endif

#if USE_ASYNC_LDS
typedef __attribute__((address_space(1))) int gint_t;   // global int*
typedef __attribute__((address_space(3))) int lint_t;   // LDS int*
// Generic "shared" pointer on amdgcn = {SHARED aperture | 32-bit LDS offset};
// low 32 bits are the LDS byte address, which is what the AS(3) operand needs.
__device__ __forceinline__ lint_t* lds_cast(void* p) {
    return (lint_t*)(uint32_t)(uintptr_t)p;
}
__device__ __forceinline__ gint_t* glb_cast(const void* p) {
    return (gint_t*)(uintptr_t)p;
}
#endif

// jnp.pad(mode='reflect') index map (single reflection; max overhang 8 << 512)
__device__ __forceinline__ int refl(int i, int n) {
    return (i < 0) ? -i : ((i >= n) ? (2 * n - 2 - i) : i);
}

__global__ __launch_bounds__(NT)
void imagenorm_fused_kernel(const float* __restrict__ images,
                            const float* __restrict__ background,
                            const float* __restrict__ thr,   // (1,5,3,1,1) -> 15
                            const float* __restrict__ scl,   // (1,5,3,1,1) -> 15
                            float* __restrict__ out) {
    __shared__ float s_xs[XS_H * XS_W];   // 48x80 halo tile of x (live to the end)
    __shared__ float s_hx[HX_H * HX_W];   // 9-tap row sums of x
    __shared__ float s_res[RS_H * RS_W];  // res = x - box_mean(x) (extended tile)
    __shared__ float s_hr[HR_H * HR_W];   // 9-tap row sums of res^2

    const int tid = threadIdx.x;
    const int tx  = blockIdx.x * TILE_W;
    const int ty  = blockIdx.y * TILE_H;
    const int iz  = blockIdx.z;                 // image plane index, 0..39
    const int band = iz % NBANDS;

    const size_t HW = (size_t)IN_H * IN_W;
    const float* __restrict__ imgbase = images + (size_t)iz * HW;
    const float* __restrict__ bgbase  = background + (size_t)iz * HW;

    // per-band asinh params (broadcast scalar loads, constant-cache friendly)
    const float th0 = thr[band * 3 + 0];
    const float th1 = thr[band * 3 + 1];
    const float th2 = thr[band * 3 + 2];
    const float es0 = expf(scl[band * 3 + 0]);
    const float es1 = expf(scl[band * 3 + 1]);
    const float es2 = expf(scl[band * 3 + 2]);

    // ---- Phase 0: stage 48x80 halo tile of x into LDS (reflect indexing) ----
    for (int id = tid; id < XS_H * XS_W; id += NT) {
        const int r  = id / XS_W;
        const int cc = id - r * XS_W;
        const int gr = refl(ty - 8 + r,  IN_H);
        const int gc = refl(tx - 8 + cc, IN_W);
        const float* src = imgbase + (size_t)gr * IN_W + gc;
#if USE_ASYNC_LDS
        __builtin_amdgcn_global_load_async_to_lds_b32(glb_cast(src),
                                                      lds_cast(&s_xs[id]),
                                                      /*offset=*/0, /*cpol=*/0);
#else
        s_xs[id] = *src;
#endif
    }
#if USE_ASYNC_LDS
    __builtin_amdgcn_s_wait_asynccnt(0);  // my async->LDS writes are done
#endif
    __syncthreads();                      // everyone's writes are visible

    // ---- Phase 1: horizontal 9-tap sums of x: hx[48][72] ----
    for (int id = tid; id < HX_H * HX_W; id += NT) {
        const int r = id / HX_W;
        const int e = id - r * HX_W;
        const float* row = &s_xs[r * XS_W + e];
        float s = 0.f;
#pragma unroll
        for (int d = 0; d < 9; ++d) s += row[d];
        s_hx[id] = s;
    }
    __syncthreads();

    // ---- Phase 2: vertical combine -> res[40][72] = x - box_mean(x) ----
    for (int id = tid; id < RS_H * RS_W; id += NT) {
        const int er = id / RS_W;
        const int ec = id - er * RS_W;
        float m = 0.f;
#pragma unroll
        for (int d = 0; d < 9; ++d) m += s_hx[(er + d) * HX_W + ec];
        s_res[id] = s_xs[(er + 4) * XS_W + (ec + 4)] - m * INV81;
    }
    __syncthreads();

    // ---- Phase 3: horizontal 9-tap sums of res^2: hr[40][64] ----
    for (int id = tid; id < HR_H * HR_W; id += NT) {
        const int er = id / HR_W;
        const int oc = id - er * HR_W;
        const float* row = &s_res[er * RS_W + oc];
        float s = 0.f;
#pragma unroll
        for (int d = 0; d < 9; ++d) { const float v = row[d]; s += v * v; }
        s_hr[id] = s;
    }
    __syncthreads();

    // ---- Phase 4: finalize all 7 channels, 4 columns per task (b128 I/O) ----
    // 512 float4 tasks over the 32x64 tile -> 2 iterations per thread.
    for (int id = tid; id < (TILE_H * TILE_W) / 4; id += NT) {
        const int orow = id >> 4;             // row 0..31
        const int oc4  = (id & 15) << 2;      // col 0..60, step 4 (16B aligned)
        const int gr   = ty + orow;
        const int gc   = tx + oc4;
        const size_t pix = (size_t)gr * IN_W + gc;

        // chained box: variance of res over 9x9 (reflect via extended res tile)
        float4 s2v = make_float4(0.f, 0.f, 0.f, 0.f);
#pragma unroll
        for (int d = 0; d < 9; ++d) {
            const float4 h = *(const float4*)&s_hr[(orow + d) * HR_W + oc4];
            s2v.x += h.x; s2v.y += h.y; s2v.z += h.z; s2v.w += h.w;
        }
        const float4 rcv = *(const float4*)&s_res[(orow + 4) * RS_W + (oc4 + 4)];
        const float4 xv  = *(const float4*)&s_xs[(orow + 12) * XS_W + (oc4 + 12)];
        const float4 bgv = *(const float4*)&bgbase[pix];

        float4 cl, l0, l1, a0, a1, a2;
        const float* s2p = &s2v.x;  const float* rcp = &rcv.x;
        const float* xp  = &xv.x;   const float* bgp = &bgv.x;
        float* clp = &cl.x; float* l0p = &l0.x; float* l1p = &l1.x;
        float* a0p = &a0.x; float* a1p = &a1.x; float* a2p = &a2.x;
#pragma unroll
        for (int j = 0; j < 4; ++j) {
            const float x  = xp[j];
            const float bg = bgp[j];
            clp[j] = rcp[j] / fmaxf(sqrtf(s2p[j] * INV81), 1.0f);
            const float off = (x - bg) / sqrtf(bg);
            l0p[j] = logf(fmaxf(off + 1.0f, 1.0f));   // t = 0
            l1p[j] = logf(fmaxf(off,        1.0f));   // t = 1 -> (off-1)+1
            a0p[j] = asinhf((x - th0) * es0);
            a1p[j] = asinhf((x - th1) * es1);
            a2p[j] = asinhf((x - th2) * es2);
        }

        float* ob = out + (size_t)iz * NCH * HW + pix;
        *(float4*)&ob[0 * HW] = xv;
        *(float4*)&ob[1 * HW] = l0;
        *(float4*)&ob[2 * HW] = l1;
        *(float4*)&ob[3 * HW] = cl;
        *(float4*)&ob[4 * HW] = a0;
        *(float4*)&ob[5 * HW] = a1;
        *(float4*)&ob[6 * HW] = a2;
    }
}

extern "C" void kernel_launch(void* const* d_in, const int* in_sizes, int n_in,
                              void* d_out, int out_size, void* d_ws, size_t ws_size,
                              hipStream_t stream) {
    (void)in_sizes; (void)n_in; (void)out_size; (void)d_ws; (void)ws_size;
    const float* images     = (const float*)d_in[0];
    const float* background = (const float*)d_in[1];
    const float* thresholds = (const float*)d_in[2];
    const float* scales     = (const float*)d_in[3];
    float* out = (float*)d_out;

    dim3 grid(IN_W / TILE_W, IN_H / TILE_H, NIMG);  // (8, 16, 40)
    dim3 block(NT);
    imagenorm_fused_kernel<<<grid, block, 0, stream>>>(images, background,
                                                       thresholds, scales, out);
}